// MetaEmbedding_Classifier_4191888081208
// MI455X (gfx1250) — compile-verified
//
#include <hip/hip_runtime.h>
#include <hip/hip_bf16.h>
#include <math.h>

typedef __attribute__((ext_vector_type(2))) float v2f;
typedef __attribute__((ext_vector_type(4))) float v4f;
typedef __attribute__((ext_vector_type(8))) float v8f;

#define BROWS 64
#define FEAT  2048
#define NCLS  1000
#define SCALE_REACH 10.0f
#define SCALE_COS   16.0f

#define MODE_PLAIN 0
#define MODE_BIAS  1
#define MODE_DIST  2
#define MODE_COS   3

#define KC 64            // K-chunk staged in LDS
#define LDSTR 68         // padded LDS row stride (floats): 16B-aligned rows, bank-shift 4

// ---------------------------------------------------------------------------
// Core: partial-out[64, N] (+ partial colss) for A[64,K] @ W[N,K]^T over one
// K-split. LDS-staged: coalesced global b128 -> padded LDS -> ds b64 feeding
// V_WMMA_F32_16X16X4_F32. Block = 128 threads = 4 waves = 64 output columns.
// ---------------------------------------------------------------------------
__device__ __forceinline__ void gemm_xwt_lds_body(
    const float* __restrict__ A,     // [64, K]
    const float* __restrict__ W,     // [N, K]
    float* __restrict__ pout,        // [nsplit, 64, N] partial sums
    float* __restrict__ pcolss,      // [nsplit, N] partial column sumsq (or nullptr)
    int N, int K, int cb, int split, int nsplit,
    float* a_sh, float* w_sh)
{
    const int tid   = threadIdx.x;
    const int lane  = tid & 31;
    const int wave  = tid >> 5;
    const int col16 = lane & 15;   // column within 16-wide tile (also A row index)
    const int khalf = lane >> 4;   // 0 -> K+{0,1}, 1 -> K+{2,3}
    const int n0    = cb * 64;
    const int col   = n0 + wave * 16 + col16;

    const int kspan = K / nsplit;
    const int k0    = split * kspan;

    v8f acc0 = {}, acc1 = {}, acc2 = {}, acc3 = {};
    float wss = 0.f;   // sumsq of this lane's W elements (one column, one K-half)

    for (int kc = 0; kc < kspan; kc += KC) {
        const int kbase = k0 + kc;
        // ---- stage A[64][KC] and W-tile[64][KC], fully coalesced b128 ----
        #pragma unroll
        for (int i = 0; i < 8; ++i) {
            int idx = (i * 128 + tid) * 4;   // flat float index into 64xKC tile
            int r   = idx >> 6;              // tile row 0..63
            int kk  = idx & 63;
            v4f av = *(const v4f*)(A + (size_t)r * K + kbase + kk);
            int wr = n0 + r; if (wr >= N) wr = N - 1;       // clamp; stores guarded
            const float* wgp = W + (size_t)wr * K + kbase + kk;
            v4f wv = *(const v4f*)(wgp);
            if (kc + KC < kspan)                             // prefetch next chunk
                __builtin_prefetch((const void*)(wgp + KC), 0, 0);
            *(v4f*)(&a_sh[r * LDSTR + kk]) = av;
            *(v4f*)(&w_sh[r * LDSTR + kk]) = wv;
        }
        __syncthreads();

        const float* ap = &a_sh[col16 * LDSTR + khalf * 2];
        const float* bp = &w_sh[(wave * 16 + col16) * LDSTR + khalf * 2];
        #pragma unroll
        for (int kk = 0; kk < KC; kk += 4) {
            v2f b  = *(const v2f*)(bp + kk);
            v2f x0 = *(const v2f*)(ap + kk);
            v2f x1 = *(const v2f*)(ap + 16 * LDSTR + kk);
            v2f x2 = *(const v2f*)(ap + 32 * LDSTR + kk);
            v2f x3 = *(const v2f*)(ap + 48 * LDSTR + kk);
            wss += b.x * b.x + b.y * b.y;
            acc0 = __builtin_amdgcn_wmma_f32_16x16x4_f32(false, x0, false, b, (short)0, acc0, false, false);
            acc1 = __builtin_amdgcn_wmma_f32_16x16x4_f32(false, x1, false, b, (short)0, acc1, false, false);
            acc2 = __builtin_amdgcn_wmma_f32_16x16x4_f32(false, x2, false, b, (short)0, acc2, false, false);
            acc3 = __builtin_amdgcn_wmma_f32_16x16x4_f32(false, x3, false, b, (short)0, acc3, false, false);
        }
        __syncthreads();
    }

    // column sumsq for this split: lane pair (c, c+16) each hold half the K range
    float colss = wss + __shfl_xor(wss, 16, 32);

    if (col < N) {
        if (pcolss != nullptr && khalf == 0)
            pcolss[(size_t)split * N + col] = colss;
        #pragma unroll
        for (int m = 0; m < 4; ++m) {
            v8f a = (m == 0) ? acc0 : (m == 1) ? acc1 : (m == 2) ? acc2 : acc3;
            #pragma unroll
            for (int v = 0; v < 8; ++v) {
                int row = m * 16 + v + khalf * 8;  // D layout: VGPR v = row v (+8 upper lanes)
                pout[((size_t)split * BROWS + row) * N + col] = a[v];
            }
        }
    }
}

// Fused stage-1: dist / hall / sel GEMMs in one grid for machine-wide occupancy.
// blocks: [0,64) dist (16 cb x 4 splits), [64,128) hall, [128,256) sel (32 cb x 4).
__global__ __launch_bounds__(128) void stage1_kernel(
    const float* __restrict__ x, const float* __restrict__ cent,
    const float* __restrict__ Whall, const float* __restrict__ Wsel,
    float* __restrict__ p_dist, float* __restrict__ p_hall,
    float* __restrict__ p_sel, float* __restrict__ p_cdist)
{
    __shared__ float a_sh[64 * LDSTR];
    __shared__ float w_sh[64 * LDSTR];
    int bid = blockIdx.x;
    if (bid < 64) {
        gemm_xwt_lds_body(x, cent, p_dist, p_cdist, NCLS, FEAT, bid & 15, bid >> 4, 4, a_sh, w_sh);
    } else if (bid < 128) {
        int b = bid - 64;
        gemm_xwt_lds_body(x, Whall, p_hall, nullptr, NCLS, FEAT, b & 15, b >> 4, 4, a_sh, w_sh);
    } else {
        int b = bid - 128;
        gemm_xwt_lds_body(x, Wsel, p_sel, nullptr, FEAT, FEAT, b & 31, b >> 5, 4, a_sh, w_sh);
    }
}

// CosNorm GEMM (same machinery, runs after `ex` is ready). 16 cb x 4 splits.
__global__ __launch_bounds__(128) void gemm_cos_kernel(
    const float* __restrict__ ex, const float* __restrict__ Wcos,
    float* __restrict__ p_cos, float* __restrict__ p_ccos)
{
    __shared__ float a_sh[64 * LDSTR];
    __shared__ float w_sh[64 * LDSTR];
    gemm_xwt_lds_body(ex, Wcos, p_cos, p_ccos, NCLS, FEAT,
                      blockIdx.x & 15, blockIdx.x >> 4, 4, a_sh, w_sh);
}

// partial-out[64,N] = A[64,K] @ B[K,N] over one K-split (B row-major; B loads
// are lane-coalesced, A is small & L2-resident -> direct loads are fine here).
__global__ __launch_bounds__(128) void gemm_ab_kernel(
    const float* __restrict__ A, const float* __restrict__ B,
    float* __restrict__ pout, int N, int K, int nsplit, int ncb)
{
    const int cb    = blockIdx.x % ncb;
    const int split = blockIdx.x / ncb;
    const int kspan = K / nsplit;
    const int k0    = split * kspan;

    const int lane  = threadIdx.x & 31;
    const int wave  = threadIdx.x >> 5;
    const int col16 = lane & 15;
    const int khalf = lane >> 4;
    const int n0    = cb * 64 + wave * 16;
    const int col   = (n0 + col16 < N) ? (n0 + col16) : (N - 1);

    const float* a0 = A + (size_t)(0 * 16 + col16) * K + khalf * 2 + k0;
    const float* a1 = A + (size_t)(1 * 16 + col16) * K + khalf * 2 + k0;
    const float* a2 = A + (size_t)(2 * 16 + col16) * K + khalf * 2 + k0;
    const float* a3 = A + (size_t)(3 * 16 + col16) * K + khalf * 2 + k0;
    const float* bp = B + (size_t)(k0 + khalf * 2) * N + col;

    v8f acc0 = {}, acc1 = {}, acc2 = {}, acc3 = {};
    for (int k = 0; k < kspan; k += 4) {
        v2f b;
        b.x = bp[(size_t)k * N];
        b.y = bp[(size_t)k * N + N];
        v2f x0 = *(const v2f*)(a0 + k);
        v2f x1 = *(const v2f*)(a1 + k);
        v2f x2 = *(const v2f*)(a2 + k);
        v2f x3 = *(const v2f*)(a3 + k);
        acc0 = __builtin_amdgcn_wmma_f32_16x16x4_f32(false, x0, false, b, (short)0, acc0, false, false);
        acc1 = __builtin_amdgcn_wmma_f32_16x16x4_f32(false, x1, false, b, (short)0, acc1, false, false);
        acc2 = __builtin_amdgcn_wmma_f32_16x16x4_f32(false, x2, false, b, (short)0, acc2, false, false);
        acc3 = __builtin_amdgcn_wmma_f32_16x16x4_f32(false, x3, false, b, (short)0, acc3, false, false);
    }

    if (n0 + col16 < N) {
        #pragma unroll
        for (int m = 0; m < 4; ++m) {
            v8f a = (m == 0) ? acc0 : (m == 1) ? acc1 : (m == 2) ? acc2 : acc3;
            #pragma unroll
            for (int v = 0; v < 8; ++v) {
                int row = m * 16 + v + khalf * 8;
                pout[((size_t)split * BROWS + row) * N + col] = a[v];
            }
        }
    }
}

// Deterministic split-K reduction + mode epilogue.
__global__ __launch_bounds__(256) void reduce_kernel(
    const float* __restrict__ part,   // [nsplit, 64, N]
    const float* __restrict__ pcolss, // [nsplit, N] or nullptr
    const float* __restrict__ bias,   // [N] or nullptr
    const float* __restrict__ xn2,    // [64] or nullptr
    float* __restrict__ out,          // [64, N]
    int N, int nsplit, int mode)
{
    int idx = blockIdx.x * 256 + threadIdx.x;
    if (idx >= BROWS * N) return;
    int r = idx / N, c = idx - r * N;
    float s = 0.f;
    for (int sp = 0; sp < nsplit; ++sp)
        s += part[((size_t)sp * BROWS + r) * N + c];
    float val;
    if (mode == MODE_DIST) {
        float cs = 0.f;
        for (int sp = 0; sp < nsplit; ++sp) cs += pcolss[(size_t)sp * N + c];
        val = sqrtf(fmaxf(xn2[r] - 2.0f * s + cs, 0.0f));
    } else if (mode == MODE_COS) {
        float cs = 0.f;
        for (int sp = 0; sp < nsplit; ++sp) cs += pcolss[(size_t)sp * N + c];
        val = s * SCALE_COS * rsqrtf(fmaxf(cs, 1e-30f));
    } else if (mode == MODE_BIAS) {
        val = s + bias[c];
    } else {
        val = s;
    }
    out[idx] = val;
}

__global__ __launch_bounds__(256) void rowsumsq_kernel(const float* __restrict__ x,
                                                       float* __restrict__ outv, int K)
{
    __shared__ float red[256];
    int r = blockIdx.x;
    float ss = 0.f;
    for (int f = threadIdx.x; f < K; f += 256) {
        float v = x[(size_t)r * K + f];
        ss += v * v;
    }
    red[threadIdx.x] = ss; __syncthreads();
    for (int s = 128; s > 0; s >>= 1) {
        if (threadIdx.x < s) red[threadIdx.x] += red[threadIdx.x + s];
        __syncthreads();
    }
    if (threadIdx.x == 0) outv[r] = red[0];
}

__global__ __launch_bounds__(256) void rowmin_kernel(const float* __restrict__ dist,
                                                     float* __restrict__ reach, int C)
{
    __shared__ float red[256];
    int r = blockIdx.x;
    float m = 3.4e38f;
    for (int c = threadIdx.x; c < C; c += 256)
        m = fminf(m, dist[(size_t)r * C + c]);
    red[threadIdx.x] = m; __syncthreads();
    for (int s = 128; s > 0; s >>= 1) {
        if (threadIdx.x < s) red[threadIdx.x] = fminf(red[threadIdx.x], red[threadIdx.x + s]);
        __syncthreads();
    }
    if (threadIdx.x == 0) reach[r] = SCALE_REACH / red[0];
}

__global__ __launch_bounds__(256) void softmax_kernel(const float* __restrict__ in,
                                                      float* __restrict__ out, int C)
{
    __shared__ float red[256];
    int r = blockIdx.x;
    float lmax = -3.4e38f;
    for (int c = threadIdx.x; c < C; c += 256)
        lmax = fmaxf(lmax, in[(size_t)r * C + c]);
    red[threadIdx.x] = lmax; __syncthreads();
    for (int s = 128; s > 0; s >>= 1) {
        if (threadIdx.x < s) red[threadIdx.x] = fmaxf(red[threadIdx.x], red[threadIdx.x + s]);
        __syncthreads();
    }
    float smax = red[0]; __syncthreads();
    float lsum = 0.f;
    for (int c = threadIdx.x; c < C; c += 256) {
        float e = __expf(in[(size_t)r * C + c] - smax);
        out[(size_t)r * C + c] = e;
        lsum += e;
    }
    red[threadIdx.x] = lsum; __syncthreads();
    for (int s = 128; s > 0; s >>= 1) {
        if (threadIdx.x < s) red[threadIdx.x] += red[threadIdx.x + s];
        __syncthreads();
    }
    float inv = 1.0f / red[0];
    for (int c = threadIdx.x; c < C; c += 256)
        out[(size_t)r * C + c] *= inv;
}

__global__ __launch_bounds__(256) void fuse_kernel(
    const float* __restrict__ x, const float* __restrict__ ssel,
    const float* __restrict__ memf, const float* __restrict__ reach,
    float* __restrict__ out_direct, float* __restrict__ out_infused,
    float* __restrict__ ex)
{
    __shared__ float red[256];
    int r = blockIdx.x;
    float rc = reach[r];
    float ss = 0.f;
    for (int f = threadIdx.x; f < FEAT; f += 256) {
        size_t i = (size_t)r * FEAT + f;
        float xv  = x[i];
        float sel = tanhf(ssel[i]);
        float inf = sel * memf[i];
        float feat = rc * (xv + inf);
        out_direct[i]  = xv;
        out_infused[i] = inf;
        ex[i] = feat;
        ss += feat * feat;
    }
    red[threadIdx.x] = ss; __syncthreads();
    for (int s = 128; s > 0; s >>= 1) {
        if (threadIdx.x < s) red[threadIdx.x] += red[threadIdx.x + s];
        __syncthreads();
    }
    float inv = 1.0f / (1.0f + sqrtf(red[0]));   // ex = feat / (1 + ||feat||)
    for (int f = threadIdx.x; f < FEAT; f += 256)
        ex[(size_t)r * FEAT + f] *= inv;
}

extern "C" void kernel_launch(void* const* d_in, const int* in_sizes, int n_in,
                              void* d_out, int out_size, void* d_ws, size_t ws_size,
                              hipStream_t stream) {
    const float* x      = (const float*)d_in[0];   // [64, 2048]
    const float* cent   = (const float*)d_in[1];   // [1000, 2048]
    const float* W_hall = (const float*)d_in[2];   // [1000, 2048]
    const float* b_hall = (const float*)d_in[3];   // [1000]
    const float* W_sel  = (const float*)d_in[4];   // [2048, 2048]
    const float* b_sel  = (const float*)d_in[5];   // [2048]
    const float* W_cos  = (const float*)d_in[6];   // [1000, 2048]

    float* out = (float*)d_out;
    float* out_logits  = out;                       // [64, 1000]
    float* out_direct  = out + 64 * NCLS;           // [64, 2048]
    float* out_infused = out + 64 * NCLS + 64 * FEAT;

    // Workspace (~6.2 MB). Partial region (ws[0 .. 1040288)) is reused across
    // the three sequential GEMM stages; stream ordering makes this safe.
    float* ws      = (float*)d_ws;
    float* p_dist  = ws;                 // 4*64*1000 = 256000
    float* p_hall  = ws + 256000;        // 256000
    float* p_sel   = ws + 512000;        // 4*64*2048 = 524288
    float* p_cdist = ws + 1036288;       // 4*1000 = 4000   (region ends 1040288)
    float* p_memf  = ws;                 // reuse: 5*64*2048 = 655360
    float* p_cos   = ws;                 // reuse: 256000
    float* p_ccos  = ws + 256000;        // reuse: 4000
    float* fin     = ws + 1040288;
    float* f_dist  = fin;                // 64000
    float* f_hall  = fin + 64000;        // 64000
    float* f_vm    = fin + 128000;       // 64000
    float* f_ssel  = fin + 192000;       // 131072
    float* f_memf  = fin + 323072;       // 131072
    float* f_ex    = fin + 454144;       // 131072
    float* f_xn2   = fin + 585216;       // 64
    float* f_reach = fin + 585280;       // 64

    // 1) ||x_r||^2
    rowsumsq_kernel<<<64, 256, 0, stream>>>(x, f_xn2, FEAT);

    // 2) fused split-K GEMMs: x@centT (+||c||^2), x@W_hallT, x@W_selT  (1024 waves)
    stage1_kernel<<<256, 128, 0, stream>>>(x, cent, W_hall, W_sel,
                                           p_dist, p_hall, p_sel, p_cdist);

    // 3) split-K reductions + epilogues
    reduce_kernel<<<(BROWS * NCLS + 255) / 256, 256, 0, stream>>>(
        p_dist, p_cdist, nullptr, f_xn2, f_dist, NCLS, 4, MODE_DIST);
    reduce_kernel<<<(BROWS * NCLS + 255) / 256, 256, 0, stream>>>(
        p_hall, nullptr, b_hall, nullptr, f_hall, NCLS, 4, MODE_BIAS);
    reduce_kernel<<<(BROWS * FEAT + 255) / 256, 256, 0, stream>>>(
        p_sel, nullptr, b_sel, nullptr, f_ssel, FEAT, 4, MODE_BIAS);

    // 4) reachability = 10 / min_dist
    rowmin_kernel<<<64, 256, 0, stream>>>(f_dist, f_reach, NCLS);

    // 5) values_memory = softmax(hall)
    softmax_kernel<<<64, 256, 0, stream>>>(f_hall, f_vm, NCLS);

    // 6) memory_feature = values_memory @ centroids (split-K = 5: 200|4)
    gemm_ab_kernel<<<32 * 5, 128, 0, stream>>>(f_vm, cent, p_memf, FEAT, NCLS, 5, 32);
    reduce_kernel<<<(BROWS * FEAT + 255) / 256, 256, 0, stream>>>(
        p_memf, nullptr, nullptr, nullptr, f_memf, FEAT, 5, MODE_PLAIN);

    // 7) fuse: tanh selector, infuse, reach gate, norm-gated unit vec
    fuse_kernel<<<64, 256, 0, stream>>>(
        x, f_ssel, f_memf, f_reach, out_direct, out_infused, f_ex);

    // 8) CosNorm: logits = 16 * ex @ (W_cos/||W_cos||)^T  (row norms fused)
    gemm_cos_kernel<<<64, 128, 0, stream>>>(f_ex, W_cos, p_cos, p_ccos);
    reduce_kernel<<<(BROWS * NCLS + 255) / 256, 256, 0, stream>>>(
        p_cos, p_ccos, nullptr, nullptr, out_logits, NCLS, 4, MODE_COS);
}